// PowerGCN_103079215485
// MI455X (gfx1250) — compile-verified
//
#include <hip/hip_runtime.h>

typedef __attribute__((ext_vector_type(2))) float v2f;
typedef __attribute__((ext_vector_type(8))) float v8f;

#define HID 32

// ---------------------------------------------------------------- utilities

__global__ void cvt_edges_kernel(const long long* __restrict__ ei,
                                 int* __restrict__ srcE, int* __restrict__ dstE, int nE) {
    int e = blockIdx.x * blockDim.x + threadIdx.x;
    if (e >= nE) return;
    srcE[e] = (int)ei[e];
    dstE[e] = (int)ei[(size_t)nE + e];
}

__global__ void set_ones_kernel(float* __restrict__ deg, int n) {
    int i = blockIdx.x * blockDim.x + threadIdx.x;
    if (i < n) deg[i] = 1.0f;                    // self-loop contributes 1 to each node
}

__global__ void deg_edges_kernel(const int* __restrict__ dstE, float* __restrict__ deg, int nE) {
    int e = blockIdx.x * blockDim.x + threadIdx.x;
    if (e < nE) atomicAdd(&deg[dstE[e]], 1.0f);
}

__global__ void dinv_kernel(const float* __restrict__ deg, float* __restrict__ dinv, int n) {
    int i = blockIdx.x * blockDim.x + threadIdx.x;
    if (i >= n) return;
    float d = deg[i];
    dinv[i] = (d > 0.0f) ? rsqrtf(d) : 0.0f;
}

__global__ void norm_kernel(const int* __restrict__ srcE, const int* __restrict__ dstE,
                            const float* __restrict__ dinv, float* __restrict__ norm, int nE) {
    int e = blockIdx.x * blockDim.x + threadIdx.x;
    if (e >= nE) return;
    norm[e] = dinv[srcE[e]] * dinv[dstE[e]];
}

// ------------------------------------------------------- layer 1: x[N,2]@W1[2,32]

__global__ void mm_in2_kernel(const float* __restrict__ x, const float* __restrict__ W1,
                              float* __restrict__ out, int nN) {
    int t = blockIdx.x * blockDim.x + threadIdx.x;
    if (t >= nN * HID) return;
    int n = t >> 5, j = t & 31;
    out[t] = x[(size_t)n * 2 + 0] * W1[j] + x[(size_t)n * 2 + 1] * W1[HID + j];
}

// --------------------------------------- layer 2: h[N,32]@W2[32,32] via f32 WMMA

__global__ void mm_hid_wmma_kernel(const float* __restrict__ h, const float* __restrict__ W,
                                   float* __restrict__ out, int nN) {
    const int lane = threadIdx.x & 31;
    const int wave = threadIdx.x >> 5;
    const int tile = blockIdx.x * 8 + wave;      // 16-row tile per wave
    const int row0 = tile * 16;
    if (row0 >= nN) return;                      // wave-uniform: EXEC stays all-ones

    int mrow = row0 + (lane & 15);
    if (mrow >= nN) mrow = nN - 1;               // clamp loads; stores guarded below
    const int kg   = (lane >> 4) << 1;           // lanes 0-15 -> K+0,1 ; lanes 16-31 -> K+2,3
    const int ncol = lane & 15;
    const float* arow = h + (size_t)mrow * HID;

    v8f c0 = {};                                 // cols 0..15
    v8f c1 = {};                                 // cols 16..31
#pragma unroll
    for (int k0 = 0; k0 < HID; k0 += 4) {
        v2f a;  a.x  = arow[k0 + kg];                        a.y  = arow[k0 + kg + 1];
        v2f b0; b0.x = W[(size_t)(k0 + kg) * HID + ncol];    b0.y = W[(size_t)(k0 + kg + 1) * HID + ncol];
        v2f b1; b1.x = W[(size_t)(k0 + kg) * HID + 16 + ncol];
                b1.y = W[(size_t)(k0 + kg + 1) * HID + 16 + ncol];
        c0 = __builtin_amdgcn_wmma_f32_16x16x4_f32(false, a, false, b0, (short)0, c0, false, false);
        c1 = __builtin_amdgcn_wmma_f32_16x16x4_f32(false, a, false, b1, (short)0, c1, false, false);
    }
    // D layout: VGPR r -> M=row0+r (lanes 0-15) / M=row0+r+8 (lanes 16-31), N=lane&15
    const int mbase = row0 + ((lane >> 4) << 3);
#pragma unroll
    for (int r = 0; r < 8; ++r) {
        int m = mbase + r;
        if (m < nN) {
            out[(size_t)m * HID + ncol]      = c0[r];
            out[(size_t)m * HID + 16 + ncol] = c1[r];
        }
    }
}

// ------------------------------------------------------- layer 3: h[N,32]@w3[32,1]

__global__ void mm_out1_kernel(const float* __restrict__ h, const float* __restrict__ w3,
                               float* __restrict__ out, int nN) {
    int n = blockIdx.x * blockDim.x + threadIdx.x;
    if (n >= nN) return;
    const float* r = h + (size_t)n * HID;
    float s = 0.0f;
#pragma unroll
    for (int k = 0; k < HID; ++k) s += r[k] * w3[k];
    out[n] = s;
}

// ------------------------------------------- aggregation: self-loop init + edges

__global__ void agg_init_kernel(const float* __restrict__ dinv, const float* __restrict__ h,
                                float* __restrict__ out, int nN) {
    int t = blockIdx.x * blockDim.x + threadIdx.x;
    if (t >= nN * HID) return;
    float di = dinv[t >> 5];
    out[t] = di * di * h[t];
}

__global__ void agg_edges_kernel(const int* __restrict__ srcE, const int* __restrict__ dstE,
                                 const float* __restrict__ norm, const float* __restrict__ h,
                                 float* out, int nE) {
    int t = blockIdx.x * blockDim.x + threadIdx.x;
    int e = t >> 3;                               // 8 threads per edge, 4 floats each
    if (e >= nE) return;
    int q = (t & 7) << 2;
    int s = srcE[e], d = dstE[e];
    float w = norm[e];
    const float4 hv = *(const float4*)(h + (size_t)s * HID + q);
    float* o = out + (size_t)d * HID + q;
    atomicAdd(o + 0, w * hv.x);
    atomicAdd(o + 1, w * hv.y);
    atomicAdd(o + 2, w * hv.z);
    atomicAdd(o + 3, w * hv.w);
}

__global__ void agg1_init_kernel(const float* __restrict__ dinv, const float* __restrict__ h,
                                 float* __restrict__ out, int nN) {
    int n = blockIdx.x * blockDim.x + threadIdx.x;
    if (n >= nN) return;
    float di = dinv[n];
    out[n] = di * di * h[n];
}

__global__ void agg1_edges_kernel(const int* __restrict__ srcE, const int* __restrict__ dstE,
                                  const float* __restrict__ norm, const float* __restrict__ h,
                                  float* out, int nE) {
    int e = blockIdx.x * blockDim.x + threadIdx.x;
    if (e >= nE) return;
    atomicAdd(&out[dstE[e]], norm[e] * h[srcE[e]]);
}

// --------------------------------------------- bias + LayerNorm + ReLU (wave/node)

__global__ void bias_ln_relu_kernel(const float* __restrict__ in, const float* __restrict__ b,
                                    const float* __restrict__ g, const float* __restrict__ be,
                                    float* __restrict__ out, int nN) {
    int node = blockIdx.x * (blockDim.x >> 5) + (threadIdx.x >> 5);
    int lane = threadIdx.x & 31;
    if (node >= nN) return;
    float v = in[(size_t)node * HID + lane] + b[lane];
    float s = v;
#pragma unroll
    for (int o = 16; o > 0; o >>= 1) s += __shfl_xor(s, o, 32);
    float mu = s * (1.0f / 32.0f);
    float dv = v - mu;
    float vs = dv * dv;
#pragma unroll
    for (int o = 16; o > 0; o >>= 1) vs += __shfl_xor(vs, o, 32);
    float inv = rsqrtf(vs * (1.0f / 32.0f) + 1e-5f);
    float y = dv * inv * g[lane] + be[lane];
    out[(size_t)node * HID + lane] = fmaxf(y, 0.0f);
}

__global__ void final_bias_kernel(const float* __restrict__ o3, const float* __restrict__ b3,
                                  float* __restrict__ out, int nN) {
    int n = blockIdx.x * blockDim.x + threadIdx.x;
    if (n >= nN) return;
    out[n] = o3[n] + b3[0];
}

// -------------------------------------------------------------------- launcher

static inline int cdiv(long long a, long long b) { return (int)((a + b - 1) / b); }

extern "C" void kernel_launch(void* const* d_in, const int* in_sizes, int n_in,
                              void* d_out, int out_size, void* d_ws, size_t ws_size,
                              hipStream_t stream) {
    const float*     x   = (const float*)d_in[0];
    const long long* ei  = (const long long*)d_in[1];
    const float*     w1  = (const float*)d_in[2];
    const float*     b1  = (const float*)d_in[3];
    const float*     g1  = (const float*)d_in[4];
    const float*     be1 = (const float*)d_in[5];
    const float*     w2  = (const float*)d_in[6];
    const float*     b2  = (const float*)d_in[7];
    const float*     g2  = (const float*)d_in[8];
    const float*     be2 = (const float*)d_in[9];
    const float*     w3  = (const float*)d_in[10];
    const float*     b3  = (const float*)d_in[11];
    float* out = (float*)d_out;

    const int N = in_sizes[0] / 2;   // x is [N,2]
    const int E = in_sizes[1] / 2;   // edge_index is [2,E]

    char* ws = (char*)d_ws;
    size_t off = 0;
    auto alloc = [&](size_t bytes) -> void* {
        void* p = ws + off;
        off += (bytes + 255) & ~(size_t)255;
        return p;
    };
    int*   srcE = (int*)  alloc(sizeof(int)   * (size_t)E);
    int*   dstE = (int*)  alloc(sizeof(int)   * (size_t)E);
    float* norm = (float*)alloc(sizeof(float) * (size_t)E);
    float* deg  = (float*)alloc(sizeof(float) * (size_t)N);
    float* dinv = (float*)alloc(sizeof(float) * (size_t)N);
    float* bufA = (float*)alloc(sizeof(float) * (size_t)N * HID);
    float* bufB = (float*)alloc(sizeof(float) * (size_t)N * HID);
    float* h3   = (float*)alloc(sizeof(float) * (size_t)N);
    float* o3   = (float*)alloc(sizeof(float) * (size_t)N);
    (void)ws_size; (void)n_in; (void)out_size;

    const int B = 256;

    // --- graph preprocessing (shared by all 3 layers)
    cvt_edges_kernel<<<cdiv(E, B), B, 0, stream>>>(ei, srcE, dstE, E);
    set_ones_kernel<<<cdiv(N, B), B, 0, stream>>>(deg, N);
    deg_edges_kernel<<<cdiv(E, B), B, 0, stream>>>(dstE, deg, E);
    dinv_kernel<<<cdiv(N, B), B, 0, stream>>>(deg, dinv, N);
    norm_kernel<<<cdiv(E, B), B, 0, stream>>>(srcE, dstE, dinv, norm, E);

    // --- layer 1: h = x@W1 ; aggregate ; +b1, LN, ReLU
    mm_in2_kernel<<<cdiv((long long)N * HID, B), B, 0, stream>>>(x, w1, bufA, N);
    agg_init_kernel<<<cdiv((long long)N * HID, B), B, 0, stream>>>(dinv, bufA, bufB, N);
    agg_edges_kernel<<<cdiv((long long)E * 8, B), B, 0, stream>>>(srcE, dstE, norm, bufA, bufB, E);
    bias_ln_relu_kernel<<<cdiv(N, B / 32), B, 0, stream>>>(bufB, b1, g1, be1, bufA, N);

    // --- layer 2: h = act@W2 (WMMA f32 16x16x4) ; aggregate ; +b2, LN, ReLU
    {
        int tiles = cdiv(N, 16);
        mm_hid_wmma_kernel<<<cdiv(tiles, 8), B, 0, stream>>>(bufA, w2, bufB, N);
    }
    agg_init_kernel<<<cdiv((long long)N * HID, B), B, 0, stream>>>(dinv, bufB, bufA, N);
    agg_edges_kernel<<<cdiv((long long)E * 8, B), B, 0, stream>>>(srcE, dstE, norm, bufB, bufA, E);
    bias_ln_relu_kernel<<<cdiv(N, B / 32), B, 0, stream>>>(bufA, b2, g2, be2, bufB, N);

    // --- layer 3: h3 = act@w3 ; aggregate ; +b3 -> d_out
    mm_out1_kernel<<<cdiv(N, B), B, 0, stream>>>(bufB, w3, h3, N);
    agg1_init_kernel<<<cdiv(N, B), B, 0, stream>>>(dinv, h3, o3, N);
    agg1_edges_kernel<<<cdiv(E, B), B, 0, stream>>>(srcE, dstE, norm, h3, o3, E);
    final_bias_kernel<<<cdiv(N, B), B, 0, stream>>>(o3, b3, out, N);
}